// SmartPixelModel_36532991820367
// MI455X (gfx1250) — compile-verified
//
#include <hip/hip_runtime.h>
#include <hip/hip_bf16.h>

typedef float v2f __attribute__((ext_vector_type(2)));
typedef float v8f __attribute__((ext_vector_type(8)));

// tanh(x) = 1 - 2/(exp(2x)+1); exp(2x) = exp2(x * 2*log2(e))
__device__ __forceinline__ float tanh_fast(float x) {
    float e = __builtin_amdgcn_exp2f(x * 2.8853900817779268f);
    return 1.0f - 2.0f * __builtin_amdgcn_rcpf(e + 1.0f);
}

#define FEAT_STRIDE 93   // odd-in-dwords stride -> conflict-free feature reads
#define SCR_STRIDE  17

__global__ __launch_bounds__(128)
void SmartPixelModel_36532991820367_kernel(
    const float* __restrict__ x,       // (B,2,13,21)
    const float* __restrict__ dw_w,    // (2,1,3,3)
    const float* __restrict__ pw_w,    // (5,2,1,1)
    const float* __restrict__ pw_b,    // (5,)
    const float* __restrict__ conv_w,  // (5,5,1,1)
    const float* __restrict__ conv_b,  // (5,)
    const float* __restrict__ w1,      // (16,90)
    const float* __restrict__ b1,      // (16,)
    const float* __restrict__ w2,      // (16,16)
    const float* __restrict__ b2,      // (16,)
    const float* __restrict__ w3,      // (14,16)
    const float* __restrict__ b3,      // (14,)
    float* __restrict__ out,           // (B,14)
    int n)
{
    __shared__ float feats[4][32][FEAT_STRIDE]; // per-wave 32 samples x 92(+1) features
    __shared__ float scr[4][32][SCR_STRIDE];    // per-wave inter-layer activations

    const int lane = threadIdx.x & 31;
    const int wave = threadIdx.x >> 5;
    const int blockBase = blockIdx.x * 128 + wave * 32;
    const int sample = blockBase + lane;
    const int sclamp = sample < n ? sample : (n - 1);
    const float* xs = x + (size_t)sclamp * 546;   // 2*13*21

    // ---- preload tiny weights (uniform across threads) ----
    float dw[2][3][3], pw[5][2], pwb[5], cw[5][5], cb[5];
#pragma unroll
    for (int c = 0; c < 2; ++c)
#pragma unroll
        for (int u = 0; u < 3; ++u)
#pragma unroll
            for (int v = 0; v < 3; ++v)
                dw[c][u][v] = dw_w[c * 9 + u * 3 + v];
#pragma unroll
    for (int o = 0; o < 5; ++o) {
        pw[o][0] = pw_w[o * 2 + 0];
        pw[o][1] = pw_w[o * 2 + 1];
        pwb[o] = pw_b[o];
        cb[o] = conv_b[o];
#pragma unroll
        for (int c = 0; c < 5; ++c) cw[o][c] = conv_w[o * 5 + c];
    }

    // ================= Phase 1: conv pipeline -> 90 features (one sample/lane)
    // Crop commutes with the 1x1 convs: only compute the 9x18 region.
#pragma unroll
    for (int band = 0; band < 3; ++band) {
        float pool[5][6];
#pragma unroll
        for (int o = 0; o < 5; ++o)
#pragma unroll
            for (int p = 0; p < 6; ++p) pool[o][p] = 0.0f;

#pragma unroll
        for (int ii = 0; ii < 3; ++ii) {
            const int i = band * 3 + ii;
#pragma unroll
            for (int j = 0; j < 18; ++j) {
                float d0 = 0.0f, d1 = 0.0f;
#pragma unroll
                for (int u = 0; u < 3; ++u)
#pragma unroll
                    for (int v = 0; v < 3; ++v) {
                        d0 += xs[(i + u) * 21 + (j + v)] * dw[0][u][v];
                        d1 += xs[273 + (i + u) * 21 + (j + v)] * dw[1][u][v];
                    }
                float t[5];
#pragma unroll
                for (int o = 0; o < 5; ++o)
                    t[o] = tanh_fast(pwb[o] + pw[o][0] * d0 + pw[o][1] * d1);
#pragma unroll
                for (int o = 0; o < 5; ++o) {
                    float q = cb[o];
#pragma unroll
                    for (int c2 = 0; c2 < 5; ++c2) q += cw[o][c2] * t[c2];
                    pool[o][j / 3] += tanh_fast(q);
                }
            }
        }
#pragma unroll
        for (int o = 0; o < 5; ++o)
#pragma unroll
            for (int p = 0; p < 6; ++p)
                feats[wave][lane][o * 18 + band * 6 + p] =
                    tanh_fast(pool[o][p] * (1.0f / 9.0f));
    }
    // K padding for the 90->92 WMMA loop
    feats[wave][lane][90] = 0.0f;
    feats[wave][lane][91] = 0.0f;
    feats[wave][lane][92] = 0.0f;

    __syncthreads();

    // ================= Phase 2: MLP via fp32 WMMA (16x16x4), two 16-row groups
    const int m16 = lane & 15;
    const int hi = lane >> 4;     // 0: K sub 0/1, 1: K sub 2/3 (A/B frag layout)
    const int ksel = hi * 2;

    // ---- layer 1: (16x90) @ (90x16) + b1, K padded to 92 ----
    v8f acc[2];
    {
        const float bias1 = b1[m16];
#pragma unroll
        for (int g = 0; g < 2; ++g)
#pragma unroll
            for (int r = 0; r < 8; ++r) acc[g][r] = bias1;
    }
#pragma unroll
    for (int kk = 0; kk < 23; ++kk) {
        const int ka = kk * 4 + ksel;
        v2f bfrag;
        bfrag.x = (ka     < 90) ? w1[m16 * 90 + ka]     : 0.0f;
        bfrag.y = (ka + 1 < 90) ? w1[m16 * 90 + ka + 1] : 0.0f;
#pragma unroll
        for (int g = 0; g < 2; ++g) {
            v2f afrag;
            afrag.x = feats[wave][g * 16 + m16][ka];
            afrag.y = feats[wave][g * 16 + m16][ka + 1];
            acc[g] = __builtin_amdgcn_wmma_f32_16x16x4_f32(
                false, afrag, false, bfrag, (short)0, acc[g], false, false);
        }
    }
    // tanh + stash in D-layout: element (M = r + 8*hi, N = m16)
#pragma unroll
    for (int g = 0; g < 2; ++g)
#pragma unroll
        for (int r = 0; r < 8; ++r)
            scr[wave][g * 16 + r + 8 * hi][m16] = tanh_fast(acc[g][r]);

    __syncthreads();

    // ---- layer 2: (16x16) @ (16x16) + b2 ----
    v8f acc2[2];
    {
        const float bias2 = b2[m16];
#pragma unroll
        for (int g = 0; g < 2; ++g)
#pragma unroll
            for (int r = 0; r < 8; ++r) acc2[g][r] = bias2;
    }
#pragma unroll
    for (int kk = 0; kk < 4; ++kk) {
        const int ka = kk * 4 + ksel;
        v2f bfrag;
        bfrag.x = w2[m16 * 16 + ka];
        bfrag.y = w2[m16 * 16 + ka + 1];
#pragma unroll
        for (int g = 0; g < 2; ++g) {
            v2f afrag;
            afrag.x = scr[wave][g * 16 + m16][ka];
            afrag.y = scr[wave][g * 16 + m16][ka + 1];
            acc2[g] = __builtin_amdgcn_wmma_f32_16x16x4_f32(
                false, afrag, false, bfrag, (short)0, acc2[g], false, false);
        }
    }

    __syncthreads();   // all layer-2 reads done before overwrite
#pragma unroll
    for (int g = 0; g < 2; ++g)
#pragma unroll
        for (int r = 0; r < 8; ++r)
            scr[wave][g * 16 + r + 8 * hi][m16] = tanh_fast(acc2[g][r]);
    __syncthreads();

    // ---- layer 3: (16x16) @ (16x14->16) + b3, no activation ----
    const bool nvalid = (m16 < 14);
    const int mc = nvalid ? m16 : 0;   // clamp to stay in-bounds
    v8f acc3[2];
    {
        const float bias3 = nvalid ? b3[mc] : 0.0f;
#pragma unroll
        for (int g = 0; g < 2; ++g)
#pragma unroll
            for (int r = 0; r < 8; ++r) acc3[g][r] = bias3;
    }
#pragma unroll
    for (int kk = 0; kk < 4; ++kk) {
        const int ka = kk * 4 + ksel;
        v2f bfrag;
        bfrag.x = nvalid ? w3[mc * 16 + ka]     : 0.0f;
        bfrag.y = nvalid ? w3[mc * 16 + ka + 1] : 0.0f;
#pragma unroll
        for (int g = 0; g < 2; ++g) {
            v2f afrag;
            afrag.x = scr[wave][g * 16 + m16][ka];
            afrag.y = scr[wave][g * 16 + m16][ka + 1];
            acc3[g] = __builtin_amdgcn_wmma_f32_16x16x4_f32(
                false, afrag, false, bfrag, (short)0, acc3[g], false, false);
        }
    }

    // ---- store (B,14): D element (M = r + 8*hi, N = m16) ----
    if (nvalid) {
#pragma unroll
        for (int g = 0; g < 2; ++g)
#pragma unroll
            for (int r = 0; r < 8; ++r) {
                const int srow = blockBase + g * 16 + r + 8 * hi;
                if (srow < n) out[(size_t)srow * 14 + m16] = acc3[g][r];
            }
    }
}

extern "C" void kernel_launch(void* const* d_in, const int* in_sizes, int n_in,
                              void* d_out, int out_size, void* d_ws, size_t ws_size,
                              hipStream_t stream) {
    (void)n_in; (void)out_size; (void)d_ws; (void)ws_size;
    const float* x      = (const float*)d_in[0];
    const float* dw_w   = (const float*)d_in[1];
    const float* pw_w   = (const float*)d_in[2];
    const float* pw_b   = (const float*)d_in[3];
    const float* conv_w = (const float*)d_in[4];
    const float* conv_b = (const float*)d_in[5];
    const float* w1     = (const float*)d_in[6];
    const float* b1     = (const float*)d_in[7];
    const float* w2     = (const float*)d_in[8];
    const float* b2     = (const float*)d_in[9];
    const float* w3     = (const float*)d_in[10];
    const float* b3     = (const float*)d_in[11];
    float* out = (float*)d_out;

    const int n = in_sizes[0] / 546;       // B (= 131072)
    const int blocks = (n + 127) / 128;    // 128 samples per block (4 waves x 32)
    SmartPixelModel_36532991820367_kernel<<<blocks, 128, 0, stream>>>(
        x, dw_w, pw_w, pw_b, conv_w, conv_b, w1, b1, w2, b2, w3, b3, out, n);
}